// EfficientGlobalPointer_22351009808818
// MI455X (gfx1250) — compile-verified
//
#include <hip/hip_runtime.h>
#include <hip/hip_bf16.h>

typedef __attribute__((ext_vector_type(2))) float v2f;
typedef __attribute__((ext_vector_type(8))) float v8f;

#define B_   8
#define L_   512
#define DIN_ 1024
#define DH_  128    // 2*HEAD_SIZE
#define HS_  64
#define NL_  16
#define INF_ 1.0e12f

// -------- Kernel 1: h = X@W1 + b1 ; RoPE -> qw,kw ; bias = (h@W2 + b2)/2 --------
// One block handles a 16-row strip of X (rows row0..row0+15). 8 waves each own a
// 16-column tile of h (128 cols total). fp32 WMMA 16x16x4, K chunked through LDS.
__global__ __launch_bounds__(256) void egp_proj_rope_bias(
    const float* __restrict__ X, const float* __restrict__ W1, const float* __restrict__ b1,
    const float* __restrict__ W2, const float* __restrict__ b2,
    float* __restrict__ qw, float* __restrict__ kw, float* __restrict__ biasw)
{
  __shared__ float xs[16][132];   // 128-wide K chunk of X strip (+4 pad vs banks)
  __shared__ float hs[16][132];   // full 16x128 h strip

  const int tid  = threadIdx.x;
  const int wave = tid >> 5;
  const int lane = tid & 31;
  const int row0 = blockIdx.x * 16;
  const int n0   = wave * 16;          // this wave's h column tile
  const int ro   = lane & 15;          // M (A) / N (B) index within tile
  const int ko   = (lane >> 4) << 1;   // K split: lanes 16-31 carry K+2,K+3

  v8f acc = {};
  for (int kc = 0; kc < DIN_; kc += 128) {
    __syncthreads();
    for (int t = tid; t < 16 * 128; t += 256) {
      int r = t >> 7, c = t & 127;
      xs[r][c] = X[(size_t)(row0 + r) * DIN_ + kc + c];
    }
    __syncthreads();
    for (int k = 0; k < 128; k += 4) {
      v2f a, bf;
      // A 16x4 f32 layout: lane<16 -> {K,K+1}, lane>=16 -> {K+2,K+3}, M=lane&15
      a.x = xs[ro][k + ko];
      a.y = xs[ro][k + ko + 1];
      // B 4x16 f32 layout: v0 = row K (lanes<16) / K+2 (lanes>=16), v1 = K+1 / K+3
      bf.x = W1[(size_t)(kc + k + ko)     * DH_ + n0 + ro];
      bf.y = W1[(size_t)(kc + k + ko + 1) * DH_ + n0 + ro];
      acc = __builtin_amdgcn_wmma_f32_16x16x4_f32(false, a, false, bf,
                                                  (short)0, acc, false, false);
    }
  }

  // Spill D tile (+b1) into the shared h strip. C/D layout: VGPR g -> M = g (+8 for hi half).
  {
    const int rbase = (lane >> 4) << 3;
    for (int g = 0; g < 8; ++g)
      hs[rbase + g][n0 + ro] = acc[g] + b1[n0 + ro];
  }
  __syncthreads();

  // RoPE: qw[d]=h[2d], kw[d]=h[2d+1]; pair j rotates (2j,2j+1) with angle pos*10000^(-j/32).
  for (int t = tid; t < 16 * 32; t += 256) {
    int r = t >> 5, j = t & 31;
    double ang = (double)((row0 + r) & (L_ - 1)) * exp(-0.28782313662425575 * (double)j);
    float c = (float)cos(ang), s = (float)sin(ang);
    float q0 = hs[r][4 * j],     q1 = hs[r][4 * j + 2];
    float k0 = hs[r][4 * j + 1], k1 = hs[r][4 * j + 3];
    size_t base = (size_t)(row0 + r) * HS_;
    qw[base + 2 * j]     = q0 * c - q1 * s;
    qw[base + 2 * j + 1] = q1 * c + q0 * s;
    kw[base + 2 * j]     = k0 * c - k1 * s;
    kw[base + 2 * j + 1] = k1 * c + k0 * s;
  }

  // bias = 0.5*(h@W2 + b2): 16x32 outputs, 128-deep dots from LDS (tiny).
  for (int t = tid; t < 16 * 32; t += 256) {
    int r = t >> 5, c = t & 31;
    float a2 = b2[c];
    for (int d = 0; d < DH_; ++d) a2 += hs[r][d] * W2[(size_t)d * 32 + c];
    biasw[(size_t)(row0 + r) * 32 + c] = 0.5f * a2;
  }
}

// -------- Kernel 2: S = qw@kw^T / 8 ; out[b,l,m,n] = S + be[n,l] + bo[m,l] - masks --------
// Block: one batch, one 16-row m-tile, 8 waves = 8 n-tiles. One 16x16 WMMA tile per
// wave (K=64 -> 16 v_wmma_f32_16x16x4_f32), reused across all 16 labels in epilogue.
__global__ __launch_bounds__(256) void egp_logits(
    const float* __restrict__ qw, const float* __restrict__ kw,
    const float* __restrict__ biasw, const int* __restrict__ am,
    float* __restrict__ out)
{
  __shared__ float qs[16][68];        // +4 pad: per-lane fragment gathers hit distinct banks
  __shared__ float ks[8][16][68];

  const int tid    = threadIdx.x;
  const int wave   = tid >> 5;
  const int lane   = tid & 31;
  const int b      = blockIdx.z;
  const int m0     = blockIdx.y * 16;
  const int ntile0 = blockIdx.x * 8;
  const int n0     = (ntile0 + wave) * 16;

  for (int t = tid; t < 16 * 64; t += 256) {
    int r = t >> 6, c = t & 63;
    qs[r][c] = qw[((size_t)b * L_ + m0 + r) * HS_ + c];
  }
  for (int t = tid; t < 8 * 16 * 64; t += 256) {
    int w = t >> 10, idx = t & 1023;
    int r = idx >> 6, c = idx & 63;
    ks[w][r][c] = kw[((size_t)b * L_ + (ntile0 + w) * 16 + r) * HS_ + c];
  }
  __syncthreads();

  const int nlo = lane & 15;
  const int ko  = (lane >> 4) << 1;
  v8f acc = {};
  for (int k = 0; k < HS_; k += 4) {
    v2f a, bf;
    a.x  = qs[nlo][k + ko];        a.y  = qs[nlo][k + ko + 1];
    bf.x = ks[wave][nlo][k + ko];  bf.y = ks[wave][nlo][k + ko + 1];  // B[k][n] = kw[n][k]
    acc = __builtin_amdgcn_wmma_f32_16x16x4_f32(false, a, false, bf,
                                                (short)0, acc, false, false);
  }

  const int n     = n0 + nlo;
  const int mbase = m0 + ((lane >> 4) << 3);
  const int amn   = am[(size_t)b * L_ + n];
  float sv[8];
  for (int g = 0; g < 8; ++g) {
    int m = mbase + g;
    float pen = 0.f;
    if (am[(size_t)b * L_ + m] * amn == 0) pen += INF_;  // attention mask
    if (m > n)                             pen += INF_;  // strict causal tril(k=-1)
    sv[g] = acc[g] * 0.125f - pen;                       // 1/sqrt(HEAD_SIZE)
  }

  const float* bb = biasw + (size_t)b * L_ * 32;
  for (int l = 0; l < NL_; ++l) {
    float  be = bb[(size_t)n * 32 + 2 * l];              // even channel, varies with n
    size_t ob = (size_t)(b * NL_ + l) * L_ * L_;
    for (int g = 0; g < 8; ++g) {
      int   m  = mbase + g;
      float bo = bb[(size_t)m * 32 + 2 * l + 1];         // odd channel, varies with m
      out[ob + (size_t)m * L_ + n] = sv[g] + be + bo;    // lanes 0-15 coalesced in n
    }
  }
}

extern "C" void kernel_launch(void* const* d_in, const int* in_sizes, int n_in,
                              void* d_out, int out_size, void* d_ws, size_t ws_size,
                              hipStream_t stream) {
  (void)in_sizes; (void)n_in; (void)out_size; (void)ws_size;
  const float* X  = (const float*)d_in[0];
  const int*   am = (const int*)d_in[1];
  const float* W1 = (const float*)d_in[2];
  const float* b1 = (const float*)d_in[3];
  const float* W2 = (const float*)d_in[4];
  const float* b2 = (const float*)d_in[5];
  float* out = (float*)d_out;

  float* qw    = (float*)d_ws;                       // B*L*64 f32 (1 MiB)
  float* kw    = qw + (size_t)B_ * L_ * HS_;         // B*L*64 f32 (1 MiB)
  float* biasw = kw + (size_t)B_ * L_ * HS_;         // B*L*32 f32 (0.5 MiB)

  egp_proj_rope_bias<<<dim3(B_ * L_ / 16), dim3(256), 0, stream>>>(
      X, W1, b1, W2, b2, qw, kw, biasw);
  egp_logits<<<dim3(4, L_ / 16, B_), dim3(256), 0, stream>>>(
      qw, kw, biasw, am, out);
}